// HungarianMatcher_65747359367844
// MI455X (gfx1250) — compile-verified
//
#include <hip/hip_runtime.h>
#include <math.h>

#define BS 32
#define NQ 900
#define NT 100
#define HB 512   // hungarian block threads

typedef float v2f __attribute__((ext_vector_type(2)));
typedef float v8f __attribute__((ext_vector_type(8)));

__device__ __forceinline__ float clampf(float x, float lo, float hi) {
    return fminf(fmaxf(x, lo), hi);
}

// ---------------------------------------------------------------------------
// Kernel 1: cost-matrix tiles. One wave per 16x16 (q,t) tile.
// VALU computes 5*L1 - 2*GIoU directly in the WMMA C-fragment layout; the
// rank-2 class term (softplus(l), -l) x (1, y) is added with one
// v_wmma_f32_16x16x4_f32.
// ---------------------------------------------------------------------------
__global__ __launch_bounds__(32)
void cost_tile_kernel(const float* __restrict__ logits,
                      const float* __restrict__ boxes,
                      const int*   __restrict__ labels,
                      const float* __restrict__ tboxes,
                      float*       __restrict__ Cout)
{
    const int b    = blockIdx.z;
    const int q0   = blockIdx.y * 16;
    const int t0   = blockIdx.x * 16;
    const int lane = threadIdx.x & 31;
    const int half = lane >> 4;      // 0: lanes 0-15, 1: lanes 16-31
    const int n    = lane & 15;      // column within tile

    const float* pb = boxes  + (size_t)b * NQ * 4;
    const float* tb = tboxes + (size_t)b * NT * 4;

    const int  t    = t0 + n;
    const bool tval = (t < NT);
    float tcx = 0.f, tcy = 0.f, tw = 0.f, th = 0.f;
    if (tval) { tcx = tb[t*4+0]; tcy = tb[t*4+1]; tw = tb[t*4+2]; th = tb[t*4+3]; }
    const float tx0 = tcx - 0.5f*tw, ty0 = tcy - 0.5f*th;
    const float tx1 = tcx + 0.5f*tw, ty1 = tcy + 0.5f*th;
    const float tarea = fmaxf(tx1 - tx0, 0.f) * fmaxf(ty1 - ty0, 0.f);

    // C fragment: 5*L1 - 2*giou; row M = vr + 8*half, col N = n.
    v8f c;
    #pragma unroll
    for (int vr = 0; vr < 8; ++vr) {
        const int q = q0 + vr + half * 8;
        float val = 0.0f;
        if (q < NQ && tval) {
            const float cx = pb[q*4+0], cy = pb[q*4+1], w = pb[q*4+2], h = pb[q*4+3];
            const float l1 = fabsf(cx-tcx) + fabsf(cy-tcy) + fabsf(w-tw) + fabsf(h-th);
            const float x0 = cx - 0.5f*w, y0 = cy - 0.5f*h;
            const float x1 = cx + 0.5f*w, y1 = cy + 0.5f*h;
            const float ix0 = fmaxf(x0, tx0), iy0 = fmaxf(y0, ty0);
            const float ix1 = fminf(x1, tx1), iy1 = fminf(y1, ty1);
            const float inter = fmaxf(ix1 - ix0, 0.f) * fmaxf(iy1 - iy0, 0.f);
            const float area1 = fmaxf(x1 - x0, 0.f) * fmaxf(y1 - y0, 0.f);
            const float uni   = area1 + tarea - inter;
            const float iou   = inter / uni;
            const float ex0 = fminf(x0, tx0), ey0 = fminf(y0, ty0);
            const float ex1 = fmaxf(x1, tx1), ey1 = fmaxf(y1, ty1);
            const float carea = fmaxf(ex1 - ex0, 0.f) * fmaxf(ey1 - ey0, 0.f);
            const float giou  = iou - (carea - uni) / carea;
            val = 5.0f * l1 - 2.0f * giou;
        }
        c[vr] = val;
    }

    // A fragment (16x4 f32): lanes 0-15 hold row M=lane, K=0 in v0, K=1 in v1;
    // lanes 16-31 hold K=2/3 -> zero.
    v2f a; a.x = 0.f; a.y = 0.f;
    if (half == 0) {
        const int q = q0 + n;
        if (q < NQ) {
            const float l = logits[(size_t)b * NQ + q];
            const float p = clampf(1.0f / (1.0f + expf(-l)), 1e-8f, 1.0f - 1e-8f);
            a.x = -logf(1.0f - p);                 // softplus(l) term (coef of 1)
            a.y = -(logf(p) - logf(1.0f - p));     // -l term (coef of y_t)
        }
    }
    // B fragment (4x16 f32): lanes 0-15 col N=lane, v0 = row K=0 (ones),
    // v1 = row K=1 (labels); lanes 16-31 hold K=2/3 -> zero.
    v2f bf; bf.x = 0.f; bf.y = 0.f;
    if (half == 0) {
        bf.x = 1.0f;
        bf.y = tval ? (float)labels[(size_t)b * NT + t] : 0.0f;
    }

    // D = A x B + C  (adds the class cost, COST_CLASS = 1.0)
    v8f d = __builtin_amdgcn_wmma_f32_16x16x4_f32(false, a, false, bf,
                                                  (short)0, c, false, false);

    #pragma unroll
    for (int vr = 0; vr < 8; ++vr) {
        const int q = q0 + vr + half * 8;
        if (q < NQ && tval)
            Cout[((size_t)b * NQ + q) * NT + t] = d[vr];
    }
}

// ---------------------------------------------------------------------------
// Kernel 2: Jonker-Volgenant shortest augmenting path on cost^T (100 x 900),
// one block per batch element, column-parallel inner loops in LDS.
// Mirrors the numpy reference including first-min argmin tie-breaking.
// ---------------------------------------------------------------------------
__global__ __launch_bounds__(HB)
void hungarian_kernel(const float* __restrict__ C, float* __restrict__ outIdx)
{
    const int b = blockIdx.x;
    const float* cost = C + (size_t)b * NQ * NT;   // costT(i,j) = cost[j*NT + i]
    const int tid  = threadIdx.x;
    const int nthr = HB;
    const int n = NT, m = NQ;
    const float INF = 1e30f;

    __shared__ float u[NT + 1];
    __shared__ float v[NQ + 1];
    __shared__ float minv[NQ + 1];
    __shared__ int   way[NQ + 1];
    __shared__ int   p[NQ + 1];
    __shared__ unsigned char used[NQ + 1];
    __shared__ float rval[HB];
    __shared__ int   ridx[HB];
    __shared__ int   s_j0;
    __shared__ int   s_done;
    __shared__ int   rowcol[NT];

    for (int j = tid; j <= m; j += nthr) { v[j] = 0.0f; p[j] = 0; }
    for (int i = tid; i <= n; i += nthr) u[i] = 0.0f;
    __syncthreads();

    for (int i = 1; i <= n; ++i) {
        if (tid == 0) { p[0] = i; s_j0 = 0; }
        for (int j = tid; j <= m; j += nthr) { minv[j] = INF; way[j] = 0; used[j] = 0; }
        __syncthreads();

        while (true) {
            const int j0 = s_j0;
            if (tid == 0) used[j0] = 1;
            __syncthreads();
            const int   i0  = p[j0];
            const float ui0 = u[i0];

            // relax free columns against row i0, track local argmin
            float bestv = INF; int bestj = m + 1;
            for (int j = tid + 1; j <= m; j += nthr) {
                if (!used[j]) {
                    const float cur = cost[(size_t)(j - 1) * NT + (i0 - 1)] - ui0 - v[j];
                    if (cur < minv[j]) { minv[j] = cur; way[j] = j0; }
                    const float mv = minv[j];
                    if (mv < bestv || (mv == bestv && j < bestj)) { bestv = mv; bestj = j; }
                }
            }
            rval[tid] = bestv; ridx[tid] = bestj;
            __syncthreads();
            for (int s = nthr >> 1; s > 0; s >>= 1) {
                if (tid < s) {
                    const float ov = rval[tid + s]; const int oj = ridx[tid + s];
                    if (ov < rval[tid] || (ov == rval[tid] && oj < ridx[tid])) {
                        rval[tid] = ov; ridx[tid] = oj;
                    }
                }
                __syncthreads();
            }
            const int   j1    = ridx[0];
            const float delta = rval[0];

            // apply delta (rows on path are distinct -> no u[] conflicts)
            for (int j = tid; j <= m; j += nthr) {
                if (used[j]) { u[p[j]] += delta; v[j] -= delta; }
                else         { minv[j] -= delta; }
            }
            __syncthreads();
            if (tid == 0) { s_j0 = j1; s_done = (p[j1] == 0); }
            __syncthreads();
            if (s_done) break;
        }
        // augment (serial pointer chase)
        if (tid == 0) {
            int j0 = s_j0;
            while (j0) { const int j1 = way[j0]; p[j0] = p[j1]; j0 = j1; }
        }
        __syncthreads();
    }

    // row_to_col: column (pred) matched to each row (target)
    for (int j = tid + 1; j <= m; j += nthr) {
        const int pj = p[j];
        if (pj != 0) rowcol[pj - 1] = j - 1;
    }
    __syncthreads();

    // idx[b,0] = sorted pred indices; idx[b,1] = argsort order (target ids)
    float* o = outIdx + (size_t)b * 2 * NT;
    for (int t = tid; t < n; t += nthr) {
        const int myc = rowcol[t];
        int r = 0;
        for (int s = 0; s < n; ++s) r += (rowcol[s] < myc) ? 1 : 0;
        o[r]      = (float)myc;
        o[NT + r] = (float)t;
    }
}

// ---------------------------------------------------------------------------
extern "C" void kernel_launch(void* const* d_in, const int* in_sizes, int n_in,
                              void* d_out, int out_size, void* d_ws, size_t ws_size,
                              hipStream_t stream)
{
    (void)in_sizes; (void)n_in; (void)out_size; (void)d_ws; (void)ws_size;
    const float* logits = (const float*)d_in[0];   // [BS,Q,1]
    const float* boxes  = (const float*)d_in[1];   // [BS,Q,4]
    const int*   labels = (const int*)  d_in[2];   // [BS,T]
    const float* tboxes = (const float*)d_in[3];   // [BS,T,4]

    float* out  = (float*)d_out;                   // idx [BS,2,T] then C [BS,Q,T]
    float* Cout = out + (size_t)BS * 2 * NT;

    dim3 grid((NT + 15) / 16, (NQ + 15) / 16, BS);
    cost_tile_kernel<<<grid, 32, 0, stream>>>(logits, boxes, labels, tboxes, Cout);
    hungarian_kernel<<<BS, HB, 0, stream>>>(Cout, out);
}